// HKLut_87454124081238
// MI455X (gfx1250) — compile-verified
//
#include <hip/hip_runtime.h>

#define H_IN 1080
#define W_IN 1920
#define OW   (W_IN * 2)
#define OH   (H_IN * 2)
#define TILE_W 32
#define TILE_H 8
#define HALO 2
#define SY_W (TILE_W + 2 * HALO)   // 36
#define SY_H (TILE_H + 2 * HALO)   // 12
#define SC_W (TILE_W + 2)          // 34
#define SC_H (TILE_H + 2)          // 10
#define NTHREADS (TILE_W * TILE_H) // 256

typedef unsigned int u32x4 __attribute__((ext_vector_type(4)));
typedef int          i32x8 __attribute__((ext_vector_type(8)));
typedef int          i32x4 __attribute__((ext_vector_type(4)));

// --- CDNA5 Tensor Data Mover availability (device pass only) ---
#if defined(__has_builtin)
#  if __has_builtin(__builtin_amdgcn_tensor_load_to_lds)
#    define HK_TDM 1
#  endif
#endif
#ifndef HK_TDM
#  define HK_TDM 0
#endif

#if HK_TDM
// DMA one 16x16x2x2 f32 LUT (1024 floats = 4KB, contiguous) from global to LDS.
// D# group0: count=1, lds_addr, global_addr[56:0], type=2.
// D# group1: data_size=4B(2), tensor_dim0=1024, tensor_dim1=1, tile_dim0=1024,
//            tile_dim1=1, tensor_dim0_stride=1024. Groups 2/3 zero (<=2D tensor).
// This toolchain's builtin takes 6 args (4th/5th descriptor groups + cpol).
__device__ __forceinline__ void tdm_load_table(const float* g, unsigned lds_off) {
  unsigned long long ga = (unsigned long long)(size_t)g;
  u32x4 g0 = { 1u,                                   // count=1, user mode
               lds_off,                              // lds_addr (bytes)
               (unsigned)(ga & 0xFFFFFFFFull),       // global_addr[31:0]
               (unsigned)((ga >> 32) & 0x1FFFFFFull) // global_addr[56:32]
                 | (2u << 30) };                     // type=2 ("image")
  i32x8 g1 = { (int)(2u << 16),      // workgroup_mask=0, data_size=2 (4B)
               (int)(1024u << 16),   // tensor_dim0[15:0] at bits 63:48
               (int)(1u << 16),      // tensor_dim1=1 at bits 95:80
               (int)(1024u << 16),   // tile_dim0=1024 at bits 127:112
               1,                    // tile_dim1=1 (bits 143:128)
               1024,                 // tensor_dim0_stride (bits 207:160)
               0, 0 };
  i32x4 gz4 = { 0, 0, 0, 0 };
  i32x8 gz8 = { 0, 0, 0, 0, 0, 0, 0, 0 };
  __builtin_amdgcn_tensor_load_to_lds(g0, g1, gz4, gz4, gz8, 0);
}
#endif

// Accumulate one (offset, table) pair's 4-rotation ensemble into the 2x2 block.
// float4 layout of T[c][n]: x=(0,0) y=(0,1) z=(1,0) w=(1,1).
// r0:(p,q) r1:(1-q,p) r2:(1-p,1-q) r3:(q,1-p)
__device__ __forceinline__ void lut_acc(const float4* __restrict__ T, int c,
                                        int n0, int n1, int n2, int n3,
                                        float& a00, float& a01, float& a10, float& a11) {
  float4 t0 = T[(c << 4) | n0];
  float4 t1 = T[(c << 4) | n1];
  float4 t2 = T[(c << 4) | n2];
  float4 t3 = T[(c << 4) | n3];
  a00 += (t0.x + t1.z) + (t2.w + t3.y);
  a01 += (t0.y + t1.x) + (t2.z + t3.w);
  a10 += (t0.z + t1.w) + (t2.y + t3.x);
  a11 += (t0.w + t1.y) + (t2.x + t3.z);
}

// Separable x2 bilinear (half-pixel centers): weights (0.25,0.75)/(0.75,0.25),
// edge-clamped neighbors reproduce the reference clip exactly.
__device__ __forceinline__ void bilin2x2(float mm, float m0, float mp,
                                         float om, float o0, float op,
                                         float pm, float p0, float pp,
                                         float& r00, float& r01, float& r10, float& r11) {
  float hm0 = 0.25f * mm + 0.75f * m0, hm1 = 0.75f * m0 + 0.25f * mp;
  float hc0 = 0.25f * om + 0.75f * o0, hc1 = 0.75f * o0 + 0.25f * op;
  float hp0 = 0.25f * pm + 0.75f * p0, hp1 = 0.75f * p0 + 0.25f * pp;
  r00 = 0.25f * hm0 + 0.75f * hc0;
  r01 = 0.25f * hm1 + 0.75f * hc1;
  r10 = 0.75f * hc0 + 0.25f * hp0;
  r11 = 0.75f * hc1 + 0.25f * hp1;
}

__global__ __launch_bounds__(NTHREADS) void hklut_sr2x_kernel(
    const float* __restrict__ x,
    const float* __restrict__ t_msb_h, const float* __restrict__ t_msb_d,
    const float* __restrict__ t_msb_b, const float* __restrict__ t_lsb_h,
    const float* __restrict__ t_lsb_d, float* __restrict__ out) {
  __shared__ __align__(16) float s_tab[5 * 1024];  // msb_h, msb_d, msb_b, lsb_h, lsb_d
  __shared__ float s_y[SY_H][SY_W];
  __shared__ int   s_img[SY_H][SY_W];
  __shared__ float s_cb[SC_H][SC_W];
  __shared__ float s_cr[SC_H][SC_W];

  const int tid = threadIdx.y * TILE_W + threadIdx.x;
  const int i0 = blockIdx.y * TILE_H;
  const int j0 = blockIdx.x * TILE_W;

  const float* xY  = x;
  const float* xCb = x + (size_t)H_IN * W_IN;
  const float* xCr = x + 2 * (size_t)H_IN * W_IN;

#if HK_TDM
  if (threadIdx.y == 0) {  // one wave issues the 5 TDM descriptors
    unsigned base = (unsigned)(size_t)(void*)s_tab;
    tdm_load_table(t_msb_h, base);
    tdm_load_table(t_msb_d, base + 4096u);
    tdm_load_table(t_msb_b, base + 8192u);
    tdm_load_table(t_lsb_h, base + 12288u);
    tdm_load_table(t_lsb_d, base + 16384u);
  }
#else
  for (int t = tid; t < 1024; t += NTHREADS) {
    s_tab[t]        = t_msb_h[t];
    s_tab[1024 + t] = t_msb_d[t];
    s_tab[2048 + t] = t_msb_b[t];
    s_tab[3072 + t] = t_lsb_h[t];
    s_tab[4096 + t] = t_lsb_d[t];
  }
#endif

  // Luma tile with halo 2 (edge-clamped), plus quantized index tile.
  for (int t = tid; t < SY_H * SY_W; t += NTHREADS) {
    int r = t / SY_W, c = t - r * SY_W;
    int gy = min(max(i0 - HALO + r, 0), H_IN - 1);
    int gx = min(max(j0 - HALO + c, 0), W_IN - 1);
    float v = xY[(size_t)gy * W_IN + gx];
    s_y[r][c] = v;
    s_img[r][c] = (int)(v * 255.0f);  // v >= 0: trunc == floor
  }
  // Chroma tiles with halo 1.
  for (int t = tid; t < SC_H * SC_W; t += NTHREADS) {
    int r = t / SC_W, c = t - r * SC_W;
    int gy = min(max(i0 - 1 + r, 0), H_IN - 1);
    int gx = min(max(j0 - 1 + c, 0), W_IN - 1);
    s_cb[r][c] = xCb[(size_t)gy * W_IN + gx];
    s_cr[r][c] = xCr[(size_t)gy * W_IN + gx];
  }

#if HK_TDM
  if (threadIdx.y == 0) {
#  if __has_builtin(__builtin_amdgcn_s_wait_tensorcnt)
    __builtin_amdgcn_s_wait_tensorcnt(0);
#  else
    asm volatile("s_wait_tensorcnt 0" ::: "memory");
#  endif
  }
#endif
  __syncthreads();

  const int i = i0 + threadIdx.y;
  const int j = j0 + threadIdx.x;
  const int ly = threadIdx.y + HALO, lx = threadIdx.x + HALO;
  const int lyc = threadIdx.y + 1, lxc = threadIdx.x + 1;

  const float4* Tmh = (const float4*)(s_tab);
  const float4* Tmd = (const float4*)(s_tab + 1024);
  const float4* Tmb = (const float4*)(s_tab + 2048);
  const float4* Tlh = (const float4*)(s_tab + 3072);
  const float4* Tld = (const float4*)(s_tab + 4096);

  const int ic = s_img[ly][lx];
  const int cm = (ic >> 4) & 15;
  const int cl = ic & 15;

  float a00 = 0.f, a01 = 0.f, a10 = 0.f, a11 = 0.f;
  {  // msb_h, off (0,1): nbrs (0,+1) (+1,0) (0,-1) (-1,0)
    int v0 = s_img[ly][lx + 1], v1 = s_img[ly + 1][lx];
    int v2 = s_img[ly][lx - 1], v3 = s_img[ly - 1][lx];
    lut_acc(Tmh, cm, (v0 >> 4) & 15, (v1 >> 4) & 15, (v2 >> 4) & 15, (v3 >> 4) & 15,
            a00, a01, a10, a11);
  }
  {  // msb_d, off (1,1): nbrs (+1,+1) (+1,-1) (-1,-1) (-1,+1)
    int v0 = s_img[ly + 1][lx + 1], v1 = s_img[ly + 1][lx - 1];
    int v2 = s_img[ly - 1][lx - 1], v3 = s_img[ly - 1][lx + 1];
    lut_acc(Tmd, cm, (v0 >> 4) & 15, (v1 >> 4) & 15, (v2 >> 4) & 15, (v3 >> 4) & 15,
            a00, a01, a10, a11);
  }
  {  // msb_b, off (1,2): nbrs (+1,+2) (+2,-1) (-1,-2) (-2,+1)
    int v0 = s_img[ly + 1][lx + 2], v1 = s_img[ly + 2][lx - 1];
    int v2 = s_img[ly - 1][lx - 2], v3 = s_img[ly - 2][lx + 1];
    lut_acc(Tmb, cm, (v0 >> 4) & 15, (v1 >> 4) & 15, (v2 >> 4) & 15, (v3 >> 4) & 15,
            a00, a01, a10, a11);
  }
  {  // lsb_h, off (0,1)
    int v0 = s_img[ly][lx + 1], v1 = s_img[ly + 1][lx];
    int v2 = s_img[ly][lx - 1], v3 = s_img[ly - 1][lx];
    lut_acc(Tlh, cl, v0 & 15, v1 & 15, v2 & 15, v3 & 15, a00, a01, a10, a11);
  }
  {  // lsb_d, off (1,1)
    int v0 = s_img[ly + 1][lx + 1], v1 = s_img[ly + 1][lx - 1];
    int v2 = s_img[ly - 1][lx - 1], v3 = s_img[ly - 1][lx + 1];
    lut_acc(Tld, cl, v0 & 15, v1 & 15, v2 & 15, v3 & 15, a00, a01, a10, a11);
  }

  const float sc = 1.0f / 1020.0f;  // /4 (rot avg) /255
  float sb00 = tanhf(a00 * sc), sb01 = tanhf(a01 * sc);
  float sb10 = tanhf(a10 * sc), sb11 = tanhf(a11 * sc);

  float y00, y01, y10, y11;
  bilin2x2(s_y[ly - 1][lx - 1], s_y[ly - 1][lx], s_y[ly - 1][lx + 1],
           s_y[ly][lx - 1],     s_y[ly][lx],     s_y[ly][lx + 1],
           s_y[ly + 1][lx - 1], s_y[ly + 1][lx], s_y[ly + 1][lx + 1],
           y00, y01, y10, y11);
  float b00, b01, b10, b11;
  bilin2x2(s_cb[lyc - 1][lxc - 1], s_cb[lyc - 1][lxc], s_cb[lyc - 1][lxc + 1],
           s_cb[lyc][lxc - 1],     s_cb[lyc][lxc],     s_cb[lyc][lxc + 1],
           s_cb[lyc + 1][lxc - 1], s_cb[lyc + 1][lxc], s_cb[lyc + 1][lxc + 1],
           b00, b01, b10, b11);
  float r00, r01, r10, r11;
  bilin2x2(s_cr[lyc - 1][lxc - 1], s_cr[lyc - 1][lxc], s_cr[lyc - 1][lxc + 1],
           s_cr[lyc][lxc - 1],     s_cr[lyc][lxc],     s_cr[lyc][lxc + 1],
           s_cr[lyc + 1][lxc - 1], s_cr[lyc + 1][lxc], s_cr[lyc + 1][lxc + 1],
           r00, r01, r10, r11);

  y00 = __saturatef(y00 + sb00); y01 = __saturatef(y01 + sb01);
  y10 = __saturatef(y10 + sb10); y11 = __saturatef(y11 + sb11);
  b00 = __saturatef(b00); b01 = __saturatef(b01);
  b10 = __saturatef(b10); b11 = __saturatef(b11);
  r00 = __saturatef(r00); r01 = __saturatef(r01);
  r10 = __saturatef(r10); r11 = __saturatef(r11);

  const size_t chan = (size_t)OH * OW;
  const size_t base = (size_t)(2 * i) * OW + (size_t)(2 * j);
  *(float2*)(out + base)                   = make_float2(y00, y01);
  *(float2*)(out + base + OW)              = make_float2(y10, y11);
  *(float2*)(out + chan + base)            = make_float2(b00, b01);
  *(float2*)(out + chan + base + OW)       = make_float2(b10, b11);
  *(float2*)(out + 2 * chan + base)        = make_float2(r00, r01);
  *(float2*)(out + 2 * chan + base + OW)   = make_float2(r10, r11);
}

extern "C" void kernel_launch(void* const* d_in, const int* in_sizes, int n_in,
                              void* d_out, int out_size, void* d_ws, size_t ws_size,
                              hipStream_t stream) {
  (void)in_sizes; (void)n_in; (void)out_size; (void)d_ws; (void)ws_size;
  const float* x  = (const float*)d_in[0];
  const float* mh = (const float*)d_in[1];
  const float* md = (const float*)d_in[2];
  const float* mb = (const float*)d_in[3];
  const float* lh = (const float*)d_in[4];
  const float* ld = (const float*)d_in[5];
  float* out = (float*)d_out;

  dim3 block(TILE_W, TILE_H);
  dim3 grid(W_IN / TILE_W, H_IN / TILE_H);  // 60 x 135, exact
  hklut_sr2x_kernel<<<grid, block, 0, stream>>>(x, mh, md, mb, lh, ld, out);
}